// PyG_GCN_29901562314759
// MI455X (gfx1250) — compile-verified
//
#include <hip/hip_runtime.h>
#include <hip/hip_bf16.h>
#include <stdint.h>

typedef __attribute__((ext_vector_type(2))) float v2f;
typedef __attribute__((ext_vector_type(8))) float v8f;

#define FEAT 128           // IN_FEAT == OUT_FEAT == 128
#define TILE 16

// ---------------------------------------------------------------- kernel 1
__global__ void gcn_deg_init(float* __restrict__ deg, int N) {
    int i = blockIdx.x * blockDim.x + threadIdx.x;
    if (i < N) deg[i] = 1.0f;            // self-loop contributes 1 to in-degree
}

// ---------------------------------------------------------------- kernel 2
__global__ void gcn_deg_edges(const long long* __restrict__ ei,
                              float* __restrict__ deg, int E) {
    int e = blockIdx.x * blockDim.x + threadIdx.x;
    if (e < E) {
        long long t = ei[(long long)E + e];      // col = target
        atomicAdd(&deg[t], 1.0f);
    }
}

// ---------------------------------------------------------------- kernel 3
__global__ void gcn_dinv(const float* __restrict__ deg,
                         float* __restrict__ dis, int N) {
    int i = blockIdx.x * blockDim.x + threadIdx.x;
    if (i < N) {
        float d = deg[i];
        dis[i] = (d > 0.0f) ? rsqrtf(d) : 0.0f;
    }
}

// ---------------------------------------------------------------- kernel 4
// xw = x @ W via V_WMMA_F32_16X16X4_F32.
// Block = 256 threads = 8 waves; each wave owns one 16-row tile and produces
// all 128 output columns (8 accumulator tiles).
//
// W is staged in LDS K-pair interleaved:  Wlds[(k>>1)*256 + ((2n)^xr) + (k&1)]
// with xr = 32 if the K-pair index is odd. This makes the WMMA B fragment
// {W[k][n], W[k+1][n]} one contiguous, 8B-aligned ds_load_b64 straight into
// an even-aligned VGPR pair (no v_mov shuffling), and the XOR swizzle puts
// the two lane halves (kOff=0 vs kOff=2 -> even vs odd pair) on disjoint
// LDS bank halves.
__global__ void __launch_bounds__(256)
gcn_gemm_wmma(const float* __restrict__ x, const float* __restrict__ W,
              float* __restrict__ xw, int nRowTiles) {
    __shared__ float Wlds[FEAT * FEAT];  // 64 KB of the 320 KB WGP pool

    // cooperative swizzled stage of W: 16384 floats / 256 threads = 64 each
    {
        const float4* __restrict__ Wg = (const float4*)W;
        int base = threadIdx.x * 16;     // in float4 units
#pragma unroll
        for (int i = 0; i < 16; ++i) {
            float4 v = Wg[base + i];
            int off = (base + i) * 4;    // flat float index = k*128 + n
            int k  = off >> 7;
            int n0 = off & 127;
            int rowb = ((k >> 1) << 8) + (k & 1);
            int xr = ((k >> 1) & 1) << 5;
            Wlds[rowb + ((((n0 + 0) << 1)) ^ xr)] = v.x;
            Wlds[rowb + ((((n0 + 1) << 1)) ^ xr)] = v.y;
            Wlds[rowb + ((((n0 + 2) << 1)) ^ xr)] = v.z;
            Wlds[rowb + ((((n0 + 3) << 1)) ^ xr)] = v.w;
        }
    }
    __syncthreads();

    const int wave = threadIdx.x >> 5;
    const int lane = threadIdx.x & 31;
    const int rowTile = blockIdx.x * 8 + wave;   // wave-uniform guard -> EXEC all 1s
    if (rowTile >= nRowTiles) return;

    const int row0 = rowTile * TILE;
    const bool hi = lane >= 16;
    const int l = lane & 15;
    const int kOff = hi ? 2 : 0;         // ISA A/B fragment layout for 16x16x4 f32
    const int xr   = hi ? 32 : 0;        // pair parity == lane half in this loop

    v8f acc[8];
#pragma unroll
    for (int t = 0; t < 8; ++t) acc[t] = (v8f){0,0,0,0,0,0,0,0};

    const float* __restrict__ arow = x + (long long)(row0 + l) * FEAT;

    // precompute lane-constant column indices into the swizzled LDS image
    int coff[8];
#pragma unroll
    for (int t = 0; t < 8; ++t) {
        int n = t * TILE + l;
        coff[t] = ((n << 1) ^ xr);
    }

#pragma unroll 4
    for (int k = 0; k < FEAT; k += 4) {
        v2f a;
        a.x = arow[k + kOff + 0];
        a.y = arow[k + kOff + 1];
        const float* __restrict__ wrow = &Wlds[((k + kOff) >> 1) << 8];
#pragma unroll
        for (int t = 0; t < 8; ++t) {
            v2f b = *(const v2f*)(wrow + coff[t]);   // single ds_load_b64
            acc[t] = __builtin_amdgcn_wmma_f32_16x16x4_f32(
                false, a, false, b, (short)0, acc[t], false, false);
        }
    }

    // C/D layout: VGPR r holds M=r (lanes 0-15) and M=r+8 (lanes 16-31), N=lane&15
#pragma unroll
    for (int t = 0; t < 8; ++t) {
#pragma unroll
        for (int r = 0; r < 8; ++r) {
            int m = hi ? r + 8 : r;
            xw[(long long)(row0 + m) * FEAT + t * TILE + l] = acc[t][r];
        }
    }
}

// ---------------------------------------------------------------- kernel 5
// out = xw * dis^2 + b   (self-loop term; full overwrite, establishes out)
__global__ void gcn_self_bias(const float* __restrict__ xw,
                              const float* __restrict__ dis,
                              const float* __restrict__ b,
                              float* __restrict__ out, int N) {
    long long idx = (long long)blockIdx.x * blockDim.x + threadIdx.x;
    if (idx < (long long)N * FEAT) {
        int i = (int)(idx >> 7);
        int f = (int)(idx & (FEAT - 1));
        float s = dis[i];
        out[idx] = xw[idx] * s * s + b[f];
    }
}

// ---------------------------------------------------------------- kernel 6
// one wave per edge: gather xw[row], scale by norm, scatter-add into out[col]
__global__ void __launch_bounds__(256)
gcn_scatter(const long long* __restrict__ ei,
            const float* __restrict__ xw, const float* __restrict__ dis,
            float* __restrict__ out, int E) {
    int e = blockIdx.x * 8 + (threadIdx.x >> 5);
    if (e >= E) return;
    int lane = threadIdx.x & 31;

    long long s = ei[e];                       // source
    long long t = ei[(long long)E + e];        // target
    float nrm = dis[s] * dis[t];

    const float4* __restrict__ src = (const float4*)(xw + s * FEAT);
    float* __restrict__ dst = out + t * FEAT;

    float4 v = src[lane];                      // 32 lanes x float4 = 128 floats
    int f = lane * 4;
    atomicAdd(&dst[f + 0], v.x * nrm);
    atomicAdd(&dst[f + 1], v.y * nrm);
    atomicAdd(&dst[f + 2], v.z * nrm);
    atomicAdd(&dst[f + 3], v.w * nrm);
}

// ---------------------------------------------------------------- launcher
extern "C" void kernel_launch(void* const* d_in, const int* in_sizes, int n_in,
                              void* d_out, int out_size, void* d_ws, size_t ws_size,
                              hipStream_t stream) {
    const float*     x  = (const float*)d_in[0];
    const float*     W  = (const float*)d_in[1];
    const float*     b  = (const float*)d_in[2];
    const long long* ei = (const long long*)d_in[3];

    const int N = in_sizes[0] / FEAT;      // 100000
    const int E = in_sizes[3] / 2;         // 1600000
    float* out = (float*)d_out;

    // workspace layout
    char* ws = (char*)d_ws;
    float* xw  = (float*)ws;                                   // N*FEAT floats
    float* deg = (float*)(ws + (size_t)N * FEAT * sizeof(float));
    float* dis = deg + N;

    const int nRowTiles = (N + TILE - 1) / TILE;               // 6250

    gcn_deg_init <<<(N + 255) / 256, 256, 0, stream>>>(deg, N);
    gcn_deg_edges<<<(E + 255) / 256, 256, 0, stream>>>(ei, deg, E);
    gcn_dinv     <<<(N + 255) / 256, 256, 0, stream>>>(deg, dis, N);
    gcn_gemm_wmma<<<(nRowTiles + 7) / 8, 256, 0, stream>>>(x, W, xw, nRowTiles);
    {
        long long tot = (long long)N * FEAT;
        gcn_self_bias<<<(unsigned)((tot + 255) / 256), 256, 0, stream>>>(xw, dis, b, out, N);
    }
    gcn_scatter  <<<(E + 7) / 8, 256, 0, stream>>>(ei, xw, dis, out, E);
}